// SquaredExp_575525617907
// MI455X (gfx1250) — compile-verified
//
#include <hip/hip_runtime.h>
#include <hip/hip_bf16.h>

// Shapes from the reference: B=4, N1=N2=4096, D=64, fp32 in/out.
#define BATCH 4
#define N1    4096
#define N2    4096
#define DDIM  64

typedef __attribute__((ext_vector_type(2))) float v2f;
typedef __attribute__((ext_vector_type(8))) float v8f;

// ---------------------------------------------------------------------------
// Pass 1: squared row norms for X1 (rows 0..B*N1-1) and X2 (rows B*N1..).
// ---------------------------------------------------------------------------
__global__ __launch_bounds__(256) void sqnorm_kernel(
    const float* __restrict__ X1, const float* __restrict__ X2,
    float* __restrict__ sq /* [B*N1 + B*N2] */)
{
    int r = blockIdx.x * blockDim.x + threadIdx.x;
    const int rows1 = BATCH * N1;
    const int total = rows1 + BATCH * N2;
    if (r >= total) return;
    const float* src = (r < rows1) ? (X1 + (size_t)r * DDIM)
                                   : (X2 + (size_t)(r - rows1) * DDIM);
    const float4* p = (const float4*)src;
    float s = 0.0f;
#pragma unroll
    for (int i = 0; i < DDIM / 4; ++i) {
        float4 v = p[i];
        s += v.x * v.x + v.y * v.y + v.z * v.z + v.w * v.w;
    }
    sq[r] = s;
}

// ---------------------------------------------------------------------------
// Pass 2: one wave computes a 16(M) x 64(N) output tile of one batch.
// Cross term via V_WMMA_F32_16X16X4_F32 (exact fp32), 16 K-steps over D=64.
//
// f32 WMMA operand layout (ISA 7.12.2):
//   A 16x4 : lanes 0-15 hold rows M=lane, VGPR0=K0,VGPR1=K1;
//            lanes 16-31 hold rows M=lane-16, VGPR0=K2,VGPR1=K3.
//   B 4x16 : same pattern with columns N.
//   C/D    : VGPR r holds row (r + 8*(lane>=16)), col = lane%16.
// ---------------------------------------------------------------------------
__global__ __launch_bounds__(256) void sqexp_kernel(
    const float* __restrict__ X1, const float* __restrict__ X2,
    const float* __restrict__ x1sq, const float* __restrict__ x2sq,
    const float* __restrict__ lengthscale, const float* __restrict__ prefactor,
    float* __restrict__ out)
{
    const int lane = threadIdx.x & 31;
    const int wave = threadIdx.x >> 5;
    const int waveId = blockIdx.x * 8 + wave;          // 65536 waves total

    const int b   = waveId >> 14;                      // / (256 Mtiles * 64 Ntiles)
    const int rem = waveId & 16383;
    const int m0  = (rem >> 6) << 4;                   // M tile * 16
    const int n0  = (rem & 63) << 6;                   // N tile * 64

    const int halfSel = lane >> 4;                     // 0: K{0,1}, 1: K{2,3}
    const int lid     = lane & 15;

    const float* aP  = X1 + ((size_t)b * N1 + (m0 + lid)) * DDIM + halfSel * 2;
    const float* bP0 = X2 + ((size_t)b * N2 + (n0 + lid)) * DDIM + halfSel * 2;
    const float* bP1 = bP0 + (size_t)16 * DDIM;
    const float* bP2 = bP0 + (size_t)32 * DDIM;
    const float* bP3 = bP0 + (size_t)48 * DDIM;

    v8f acc0 = {}, acc1 = {}, acc2 = {}, acc3 = {};

#pragma unroll
    for (int k = 0; k < DDIM; k += 4) {
        v2f a  = *(const v2f*)(aP  + k);
        v2f b0 = *(const v2f*)(bP0 + k);
        v2f b1 = *(const v2f*)(bP1 + k);
        v2f b2 = *(const v2f*)(bP2 + k);
        v2f b3 = *(const v2f*)(bP3 + k);
        acc0 = __builtin_amdgcn_wmma_f32_16x16x4_f32(false, a, false, b0, (short)0, acc0, false, false);
        acc1 = __builtin_amdgcn_wmma_f32_16x16x4_f32(false, a, false, b1, (short)0, acc1, false, false);
        acc2 = __builtin_amdgcn_wmma_f32_16x16x4_f32(false, a, false, b2, (short)0, acc2, false, false);
        acc3 = __builtin_amdgcn_wmma_f32_16x16x4_f32(false, a, false, b3, (short)0, acc3, false, false);
    }

    // Epilogue: out = pf^2 * exp(-0.5 * (|x1|^2 + |x2|^2 - 2*cross) / ls^2)
    const float lsv = lengthscale[0];
    const float pfv = prefactor[0];
    const float cexp = -0.5f / (lsv * lsv);
    const float pf2  = pfv * pfv;

    const float* x1s = x1sq + (size_t)b * N1;
    const float* x2s = x2sq + (size_t)b * N2;
    const float x2v0 = x2s[n0 + lid];
    const float x2v1 = x2s[n0 + 16 + lid];
    const float x2v2 = x2s[n0 + 32 + lid];
    const float x2v3 = x2s[n0 + 48 + lid];

    float* outB = out + (size_t)b * N1 * N2;

#pragma unroll
    for (int r = 0; r < 8; ++r) {
        const int row = m0 + r + halfSel * 8;
        const float x1v = x1s[row];
        float* o = outB + (size_t)row * N2 + n0 + lid;
        const float v0 = pf2 * __expf(cexp * (x1v + x2v0 - 2.0f * acc0[r]));
        const float v1 = pf2 * __expf(cexp * (x1v + x2v1 - 2.0f * acc1[r]));
        const float v2 = pf2 * __expf(cexp * (x1v + x2v2 - 2.0f * acc2[r]));
        const float v3 = pf2 * __expf(cexp * (x1v + x2v3 - 2.0f * acc3[r]));
        __builtin_nontemporal_store(v0, o);
        __builtin_nontemporal_store(v1, o + 16);
        __builtin_nontemporal_store(v2, o + 32);
        __builtin_nontemporal_store(v3, o + 48);
    }
}

extern "C" void kernel_launch(void* const* d_in, const int* in_sizes, int n_in,
                              void* d_out, int out_size, void* d_ws, size_t ws_size,
                              hipStream_t stream) {
    const float* X1 = (const float*)d_in[0];
    const float* X2 = (const float*)d_in[1];
    const float* ls = (const float*)d_in[2];
    const float* pf = (const float*)d_in[3];
    float* out = (float*)d_out;

    float* x1sq = (float*)d_ws;                 // B*N1 floats
    float* x2sq = x1sq + BATCH * N1;            // B*N2 floats (128 KB total)

    // Pass 1: norms for all B*N1 + B*N2 rows.
    {
        const int rows = BATCH * N1 + BATCH * N2;   // 32768
        const int threads = 256;
        const int blocks = (rows + threads - 1) / threads;
        sqnorm_kernel<<<blocks, threads, 0, stream>>>(X1, X2, x1sq);
    }

    // Pass 2: 65536 waves (one 16x64 tile each), 8 waves per 256-thread block.
    {
        const int totalWaves = BATCH * (N1 / 16) * (N2 / 64); // 65536
        const int blocks = totalWaves / 8;                    // 8192
        sqexp_kernel<<<blocks, 256, 0, stream>>>(X1, X2, x1sq, x2sq, ls, pf, out);
    }
}